// Seq2SeqVanilla_24068996726864
// MI455X (gfx1250) — compile-verified
//
#include <hip/hip_runtime.h>
#include <hip/hip_bf16.h>

typedef __attribute__((ext_vector_type(2))) float v2f;
typedef __attribute__((ext_vector_type(8))) float v8f;

namespace {
constexpr int B = 512;
constexpr int S = 128;
constexpr int T = 128;
constexpr int E = 128;
constexpr int H = 1024;
constexpr int V = 128;
constexpr int PAD = 0;
}

__device__ __forceinline__ v8f wmma4(v2f a, v2f b, v8f c) {
  // D = A(16x4 f32) * B(4x16 f32) + C(16x16 f32)
  return __builtin_amdgcn_wmma_f32_16x16x4_f32(false, a, false, b, (short)0, c,
                                               false, false);
}

__device__ __forceinline__ float sigmoid_f(float x) {
  return 1.0f / (1.0f + __expf(-x));
}

// One GRU time step: h_out = GRUCell(emb[tok], h_in).
// Wave tiling: each wave owns a 32(batch) x 16(hidden-col) tile of h_out
// (two 16x16 WMMA row-tiles), so the three gates' B fragments (Whh rows for
// r/z/n) are loaded once and reused by 2 WMMAs each: 5 loads -> 6 WMMAs per
// K-chunk. 4 waves per block share the same hidden-column tile j0 so their
// B-fragment streams hit WGP$/L2.
__global__ __launch_bounds__(128) void gru_step_kernel(
    const int* __restrict__ tok, int tok_stride, const float* __restrict__ emb,
    const float* __restrict__ h_in, float* __restrict__ h_out,
    const float* __restrict__ Wih, const float* __restrict__ Whh,
    const float* __restrict__ bih, const float* __restrict__ bhh) {
  const int wave = threadIdx.x >> 5;
  const int lane = threadIdx.x & 31;
  const int lm = lane & 15;            // A row (m) / B col (n)
  const int khalf = (lane >> 4) << 1;  // K sub-offset within a 4-chunk
  const int j0 = (blockIdx.x & 63) << 4;                   // 64 col tiles
  const int b0 = ((((blockIdx.x >> 6) << 2) + wave)) << 5; // 16 row groups

  v8f aR0 = {}, aR1 = {};    // i_r + h_r pre-activation
  v8f aZ0 = {}, aZ1 = {};    // i_z + h_z pre-activation
  v8f aIN0 = {}, aIN1 = {};  // i_n pre-activation
  v8f aHN0 = {}, aHN1 = {};  // h_n pre-activation (n = tanh(i_n + r*h_n))

  // --- recurrent GEMM: gh = h_in @ Whh^T, three gate column blocks ---
  const float* hrow = h_in + (size_t)(b0 + lm) * H + khalf;  // rows b0..b0+15
  // rows b0+16..b0+31 live at a fixed +16*H float offset (64KB immediate).
  const float* wr = Whh + (size_t)(j0 + lm) * H + khalf;
  const float* wz = wr + (size_t)H * H;
  const float* wn = wz + (size_t)H * H;
#pragma unroll 4
  for (int k = 0; k < H; k += 4) {
    v2f a0 = *(const v2f*)(hrow + k);
    v2f a1 = *(const v2f*)(hrow + 16 * H + k);
    v2f br = *(const v2f*)(wr + k);
    v2f bz = *(const v2f*)(wz + k);
    v2f bn = *(const v2f*)(wn + k);
    aR0 = wmma4(a0, br, aR0);
    aR1 = wmma4(a1, br, aR1);
    aZ0 = wmma4(a0, bz, aZ0);
    aZ1 = wmma4(a1, bz, aZ1);
    aHN0 = wmma4(a0, bn, aHN0);
    aHN1 = wmma4(a1, bn, aHN1);
  }

  // --- input GEMM: gi = x @ Wih^T with fused embedding gather ---
  // PAD token row is forced to zero via a lane multiplier (no branch, EXEC
  // stays all-ones as required by WMMA).
  const int t0 = tok[(size_t)(b0 + lm) * tok_stride];
  const int t1 = tok[(size_t)(b0 + 16 + lm) * tok_stride];
  const float xs0 = (t0 == PAD) ? 0.0f : 1.0f;
  const float xs1 = (t1 == PAD) ? 0.0f : 1.0f;
  const float* xrow0 = emb + (size_t)t0 * E + khalf;
  const float* xrow1 = emb + (size_t)t1 * E + khalf;
  const float* ur = Wih + (size_t)(j0 + lm) * E + khalf;
  const float* uz = ur + (size_t)H * E;
  const float* un = uz + (size_t)H * E;
#pragma unroll 4
  for (int k = 0; k < E; k += 4) {
    v2f a0 = *(const v2f*)(xrow0 + k);
    v2f a1 = *(const v2f*)(xrow1 + k);
    a0.x *= xs0;
    a0.y *= xs0;
    a1.x *= xs1;
    a1.y *= xs1;
    v2f br = *(const v2f*)(ur + k);
    v2f bz = *(const v2f*)(uz + k);
    v2f bn = *(const v2f*)(un + k);
    aR0 = wmma4(a0, br, aR0);
    aR1 = wmma4(a1, br, aR1);
    aZ0 = wmma4(a0, bz, aZ0);
    aZ1 = wmma4(a1, bz, aZ1);
    aIN0 = wmma4(a0, bn, aIN0);
    aIN1 = wmma4(a1, bn, aIN1);
  }

  // --- gate nonlinearities + state blend ---
  // C/D layout: VGPR v holds row (v + 8*(lane>=16)), col = lane&15.
  const int col = j0 + lm;
  const float biasR = bih[col] + bhh[col];
  const float biasZ = bih[H + col] + bhh[H + col];
  const float biasIN = bih[2 * H + col];
  const float biasHN = bhh[2 * H + col];
  const int row0 = b0 + ((lane >> 4) << 3);
#pragma unroll
  for (int v = 0; v < 8; ++v) {
    float r = sigmoid_f(aR0[v] + biasR);
    float z = sigmoid_f(aZ0[v] + biasZ);
    float n = tanhf(aIN0[v] + biasIN + r * (aHN0[v] + biasHN));
    float hp = h_in[(size_t)(row0 + v) * H + col];
    h_out[(size_t)(row0 + v) * H + col] = (1.0f - z) * n + z * hp;
  }
#pragma unroll
  for (int v = 0; v < 8; ++v) {
    float r = sigmoid_f(aR1[v] + biasR);
    float z = sigmoid_f(aZ1[v] + biasZ);
    float n = tanhf(aIN1[v] + biasIN + r * (aHN1[v] + biasHN));
    float hp = h_in[(size_t)(row0 + 16 + v) * H + col];
    h_out[(size_t)(row0 + 16 + v) * H + col] = (1.0f - z) * n + z * hp;
  }
}

// FC projection + greedy argmax for one decode step.
// Block = 8 waves: 16 batch rows x all 128 vocab columns. Preds go both to
// the global output slab and to LDS; 16 lanes then do a first-max argmax.
__global__ __launch_bounds__(256) void fc_argmax_kernel(
    const float* __restrict__ h, const float* __restrict__ fcW,
    const float* __restrict__ fcb, float* __restrict__ out, int t,
    int* __restrict__ tokens) {
  __shared__ float preds[16 * V];
  const int wave = threadIdx.x >> 5;
  const int lane = threadIdx.x & 31;
  const int lm = lane & 15;
  const int khalf = (lane >> 4) << 1;
  const int b0 = blockIdx.x << 4;
  const int j0 = wave << 4;

  v8f acc = {};
  const float* hrow = h + (size_t)(b0 + lm) * H + khalf;
  const float* wrow = fcW + (size_t)(j0 + lm) * H + khalf;
#pragma unroll 4
  for (int k = 0; k < H; k += 4) {
    v2f a = *(const v2f*)(hrow + k);
    v2f b = *(const v2f*)(wrow + k);
    acc = wmma4(a, b, acc);
  }

  const int col = j0 + lm;
  const float bias = fcb[col];
  const int row0 = (lane >> 4) << 3;
#pragma unroll
  for (int v = 0; v < 8; ++v) {
    float p = acc[v] + bias;
    preds[(row0 + v) * V + col] = p;
    out[(size_t)(b0 + row0 + v) * ((size_t)T * V) + (size_t)t * V + col] = p;
  }
  __syncthreads();
  if (threadIdx.x < 16) {
    const float* row = preds + threadIdx.x * V;
    int best = 0;
    float bv = row[0];
    for (int v = 1; v < V; ++v) {
      float p = row[v];
      if (p > bv) {  // strict '>' keeps first max, matching jnp.argmax
        bv = p;
        best = v;
      }
    }
    tokens[b0 + threadIdx.x] = best;
  }
}

__global__ void zero_f32(float* __restrict__ p, int n) {
  int i = blockIdx.x * blockDim.x + threadIdx.x;
  if (i < n) p[i] = 0.0f;
}

__global__ void zero_out_t0(float* __restrict__ out) {
  int i = blockIdx.x * blockDim.x + threadIdx.x;
  if (i < B * V) out[(size_t)(i / V) * ((size_t)T * V) + (i % V)] = 0.0f;
}

__global__ void init_tokens(const int* __restrict__ trg,
                            int* __restrict__ tokens) {
  int b = blockIdx.x * blockDim.x + threadIdx.x;
  if (b < B) tokens[b] = trg[(size_t)b * T];
}

extern "C" void kernel_launch(void* const* d_in, const int* in_sizes, int n_in,
                              void* d_out, int out_size, void* d_ws,
                              size_t ws_size, hipStream_t stream) {
  const int* src = (const int*)d_in[0];
  const int* trg = (const int*)d_in[1];
  const float* emb_src = (const float*)d_in[2];
  const float* emb_trg = (const float*)d_in[3];
  const float* enc_Wih = (const float*)d_in[4];
  const float* enc_Whh = (const float*)d_in[5];
  const float* enc_bih = (const float*)d_in[6];
  const float* enc_bhh = (const float*)d_in[7];
  const float* dec_Wih = (const float*)d_in[8];
  const float* dec_Whh = (const float*)d_in[9];
  const float* dec_bih = (const float*)d_in[10];
  const float* dec_bhh = (const float*)d_in[11];
  const float* fc_W = (const float*)d_in[12];
  const float* fc_b = (const float*)d_in[13];
  // d_in[14] = teacher_forcing_ratio == 0 -> greedy decode path only.
  float* out = (float*)d_out;

  float* hA = (float*)d_ws;
  float* hB = hA + (size_t)B * H;
  int* tokens = (int*)(hB + (size_t)B * H);

  zero_f32<<<(B * H + 255) / 256, 256, 0, stream>>>(hA, B * H);
  zero_out_t0<<<(B * V + 255) / 256, 256, 0, stream>>>(out);
  init_tokens<<<(B + 255) / 256, 256, 0, stream>>>(trg, tokens);

  // Encoder: 128 sequential GRU steps, h ping-pongs hA <-> hB.
  // Grid: 64 col-tiles x 4 row-group-blocks = 256 blocks, 4 waves each.
  for (int s = 0; s < S; ++s) {
    const float* hin = (s & 1) ? hB : hA;
    float* hout = (s & 1) ? hA : hB;
    gru_step_kernel<<<256, 128, 0, stream>>>(src + s, S, emb_src, hin, hout,
                                             enc_Wih, enc_Whh, enc_bih,
                                             enc_bhh);
  }
  // S=128 (even) -> final encoder hidden lands in hA.

  // Decoder: greedy feedback through tokens buffer.
  for (int t = 1; t < T; ++t) {
    int d = t - 1;
    const float* hin = (d & 1) ? hB : hA;
    float* hout = (d & 1) ? hA : hB;
    gru_step_kernel<<<256, 128, 0, stream>>>(tokens, 1, emb_trg, hin, hout,
                                             dec_Wih, dec_Whh, dec_bih,
                                             dec_bhh);
    fc_argmax_kernel<<<B / 16, 256, 0, stream>>>(hout, fc_W, fc_b, out, t,
                                                 tokens);
  }
}